// DistanceEstimator_7387343749780
// MI455X (gfx1250) — compile-verified
//
#include <hip/hip_runtime.h>

// ---------------------------------------------------------------------------
// DistanceEstimator (GINE GNN) for gfx1250.
// fp32 end-to-end; all GEMM portions (edge MLP, conv2 lin, node MLPs) use
// V_WMMA_F32_16X16X4_F32. B matrices stored in LDS as K-pairs so each WMMA
// B operand is a single aligned ds_load_b64.
// ---------------------------------------------------------------------------

typedef float v2f __attribute__((ext_vector_type(2)));
typedef float v8f __attribute__((ext_vector_type(8)));

#define TWO48M1 281474976710655.0f // 2^48 - 1

__device__ __forceinline__ v8f wmma4(v2f a, v2f b, v8f c) {
  // D(16x16,f32) = A(16x4,f32) * B(4x16,f32) + C
  return __builtin_amdgcn_wmma_f32_16x16x4_f32(
      /*neg_a=*/false, a, /*neg_b=*/false, b,
      /*c_mod=*/(short)0, c, /*reuse_a=*/false, /*reuse_b=*/false);
}

// Store W[K][C] (row-major, K even) into paired layout:
//   p[((k>>1)*C + c)*2 + (k&1)] = W[k][c]
// so {W[k][c], W[k+1][c]} is one aligned 8-byte LDS access.
__device__ __forceinline__ void stage_paired(const float* __restrict__ w,
                                             float* __restrict__ p,
                                             int K, int C, int tid, int nt) {
  for (int i = tid; i < K * C; i += nt) {
    int k = i / C, c = i - k * C;
    p[(((k >> 1) * C + c) << 1) + (k & 1)] = w[i];
  }
}

__device__ __forceinline__ v2f bpair(const float* __restrict__ p, int kq, int half,
                                     int C, int col) {
  // kq = k (multiple of 4); pair index = k/2 + half
  return *(const v2f*)&p[(((kq >> 1) + half) * C + col) << 1];
}

// --------------------------- node embedding --------------------------------
// x0 = MLP_1_32_32(clip(name / (2^48-1), 0, 1));  also writes agg1 = x0.
__global__ __launch_bounds__(256) void node_embed_kernel(
    const int* __restrict__ names,
    const float* __restrict__ w1, const float* __restrict__ b1,   // [1,32],[32]
    const float* __restrict__ w2, const float* __restrict__ b2,   // [32,32],[32]
    float* __restrict__ x0, float* __restrict__ agg1, int n) {
  __shared__ float sw1[32], sb1[32], sw2[32 * 32], sb2[32];
  for (int i = threadIdx.x; i < 32; i += 256) { sw1[i] = w1[i]; sb1[i] = b1[i]; sb2[i] = b2[i]; }
  for (int i = threadIdx.x; i < 32 * 32; i += 256) sw2[i] = w2[i];
  __syncthreads();
  int nid = blockIdx.x * 256 + threadIdx.x;
  if (nid >= n) return;
  float raw = (float)names[nid];
  float v = fminf(fmaxf(raw / TWO48M1, 0.0f), 1.0f);
  float h[32];
#pragma unroll
  for (int j = 0; j < 32; ++j) h[j] = fmaxf(v * sw1[j] + sb1[j], 0.0f);
  for (int k = 0; k < 32; ++k) {
    float acc = sb2[k];
#pragma unroll
    for (int j = 0; j < 32; ++j) acc += h[j] * sw2[j * 32 + k];
    x0[(size_t)nid * 32 + k] = acc;
    agg1[(size_t)nid * 32 + k] = acc;
  }
}

// ----------------- conv1 message + scatter (WMMA edge MLP) -----------------
// agg1[dst] += relu(x0[src] + edgeMLP(attr)),  d = 32.
// Wave handles 16 edges: hidden tile [16,32] via VALU (16 FMA/lane),
// e = h @ ew2 + eb2 via WMMA (2 col tiles x 8 k-steps), then gather/scatter.
__global__ __launch_bounds__(256) void conv1_msg_kernel(
    const int* __restrict__ src, const int* __restrict__ dst,
    const float* __restrict__ attr,
    const float* __restrict__ ew1, const float* __restrict__ eb1,
    const float* __restrict__ ew2, const float* __restrict__ eb2,
    const float* __restrict__ x0, float* __restrict__ agg1, int ne) {
  __shared__ float sw1[32], sb1[32], sb2[32];
  __shared__ float sw2p[32 * 32];        // paired [32,32]
  __shared__ float ht[8][16 * 32];
  __shared__ int sidx[8][32];
  for (int i = threadIdx.x; i < 32; i += 256) { sw1[i] = ew1[i]; sb1[i] = eb1[i]; sb2[i] = eb2[i]; }
  stage_paired(ew2, sw2p, 32, 32, threadIdx.x, 256);
  __syncthreads();

  const int lane = threadIdx.x & 31;
  const int wave = threadIdx.x >> 5;
  const int ln = lane & 15;
  const int half = lane >> 4;
  float* htile = ht[wave];
  int* idx = sidx[wave];
  const int e0 = (blockIdx.x * 8 + wave) * 16;

  if (lane < 16) {
    int e = e0 + lane; int ec = (e < ne) ? e : ne - 1;
    idx[lane] = src[ec];
    idx[16 + lane] = dst[ec];
  }
  { // hidden tile: lane covers edge ln, hidden cols [16*half, 16*half+16)
    int e = e0 + ln; int ec = (e < ne) ? e : ne - 1;
    float a = attr[ec];
#pragma unroll
    for (int c = 0; c < 16; ++c) {
      int j = half * 16 + c;
      htile[ln * 32 + j] = fmaxf(a * sw1[j] + sb1[j], 0.0f);
    }
  }
  __syncthreads();

  v8f acc[2]; // e[16 edges][32] in 2 col tiles
#pragma unroll
  for (int nt = 0; nt < 2; ++nt) {
    float bv = sb2[nt * 16 + ln];
#pragma unroll
    for (int v = 0; v < 8; ++v) acc[nt][v] = bv;
  }
#pragma unroll
  for (int k = 0; k < 32; k += 4) {
    v2f a = *(const v2f*)&htile[ln * 32 + k + 2 * half];
#pragma unroll
    for (int nt = 0; nt < 2; ++nt)
      acc[nt] = wmma4(a, bpair(sw2p, k, half, 32, nt * 16 + ln), acc[nt]);
  }
  // message: relu(x0[src] + e), scatter to agg1[dst]
#pragma unroll
  for (int nt = 0; nt < 2; ++nt)
#pragma unroll
    for (int v = 0; v < 8; ++v) {
      int m = v + 8 * half;
      int e = e0 + m;
      if (e < ne) {
        int col = nt * 16 + ln;
        float mv = fmaxf(acc[nt][v] + x0[(size_t)idx[m] * 32 + col], 0.0f);
        atomicAdd(&agg1[(size_t)idx[16 + m] * 32 + col], mv);
      }
    }
}

// ----------------------- node update MLP (WMMA) ----------------------------
// out = relu( relu(in @ W1 + b1) @ W2 + b2 )   (outer relu = post-conv relu)
// in: [n,K1], W1: [K1,64], W2: [64,64].  Wave handles a 16-row tile.
template <int K1>
__global__ __launch_bounds__(256) void node_mlp_kernel(
    const float* __restrict__ in,
    const float* __restrict__ w1, const float* __restrict__ b1,
    const float* __restrict__ w2, const float* __restrict__ b2,
    float* __restrict__ out, float* __restrict__ agg_copy, int n) {
  __shared__ float sw1p[K1 * 64];   // paired [K1,64]
  __shared__ float sw2p[64 * 64];   // paired [64,64]
  __shared__ float sb1[64], sb2[64];
  __shared__ float tile[8][16 * 64]; // per-wave A / hidden scratch
  stage_paired(w1, sw1p, K1, 64, threadIdx.x, 256);
  stage_paired(w2, sw2p, 64, 64, threadIdx.x, 256);
  for (int i = threadIdx.x; i < 64; i += 256) { sb1[i] = b1[i]; sb2[i] = b2[i]; }
  __syncthreads();

  const int lane = threadIdx.x & 31;
  const int wave = threadIdx.x >> 5;
  const int ln = lane & 15;
  const int half = lane >> 4;
  float* at = tile[wave];
  const int row0 = (blockIdx.x * 8 + wave) * 16;

  // stage A tile [16][K1] (compact stride K1), clamp OOB rows
  for (int i = lane; i < 16 * K1; i += 32) {
    int m = i / K1, k = i - m * K1;
    int r = row0 + m; if (r >= n) r = n - 1;
    at[m * K1 + k] = in[(size_t)r * K1 + k];
  }

  v8f acc[4];
  // ---- layer 1: hidden = relu(A @ W1 + b1), 4 col tiles of 16 ----
#pragma unroll
  for (int nt = 0; nt < 4; ++nt) {
    float bv = sb1[nt * 16 + ln];
#pragma unroll
    for (int v = 0; v < 8; ++v) acc[nt][v] = bv;
  }
#pragma unroll
  for (int k = 0; k < K1; k += 4) {
    v2f a = *(const v2f*)&at[ln * K1 + k + 2 * half];
#pragma unroll
    for (int nt = 0; nt < 4; ++nt)
      acc[nt] = wmma4(a, bpair(sw1p, k, half, 64, nt * 16 + ln), acc[nt]);
  }
  // relu + write hidden tile [16][64] (stride 64; A fully consumed)
#pragma unroll
  for (int nt = 0; nt < 4; ++nt)
#pragma unroll
    for (int v = 0; v < 8; ++v) {
      int m = v + 8 * half;
      at[m * 64 + nt * 16 + ln] = fmaxf(acc[nt][v], 0.0f);
    }

  // ---- layer 2: out = relu(h @ W2 + b2) ----
#pragma unroll
  for (int nt = 0; nt < 4; ++nt) {
    float bv = sb2[nt * 16 + ln];
#pragma unroll
    for (int v = 0; v < 8; ++v) acc[nt][v] = bv;
  }
#pragma unroll
  for (int k = 0; k < 64; k += 4) {
    v2f a = *(const v2f*)&at[ln * 64 + k + 2 * half];
#pragma unroll
    for (int nt = 0; nt < 4; ++nt)
      acc[nt] = wmma4(a, bpair(sw2p, k, half, 64, nt * 16 + ln), acc[nt]);
  }
#pragma unroll
  for (int nt = 0; nt < 4; ++nt)
#pragma unroll
    for (int v = 0; v < 8; ++v) {
      int m = v + 8 * half;
      int r = row0 + m;
      if (r < n) {
        float val = fmaxf(acc[nt][v], 0.0f);
        out[(size_t)r * 64 + nt * 16 + ln] = val;
        if (agg_copy) agg_copy[(size_t)r * 64 + nt * 16 + ln] = val;
      }
    }
}

// ----------------- conv2 message + scatter (full WMMA chain) ---------------
// agg2[dst] += relu(x1[src] + edgeMLP(attr) @ lin_w + lin_b), d = 64.
// Wave handles 16 edges:
//   hTile [16,32] via VALU -> e = h @ ew2 + eb2 via WMMA (16 ops)
//   -> ee = e @ lin_w + lin_b via WMMA (32 ops) -> gather/relu/scatter.
__global__ __launch_bounds__(256) void conv2_msg_kernel(
    const int* __restrict__ src, const int* __restrict__ dst,
    const float* __restrict__ attr,
    const float* __restrict__ ew1, const float* __restrict__ eb1,
    const float* __restrict__ ew2, const float* __restrict__ eb2,
    const float* __restrict__ lw, const float* __restrict__ lb, // [32,64],[64]
    const float* __restrict__ x1, float* __restrict__ agg2, int ne) {
  __shared__ float sew1[32], seb1[32], seb2[32];
  __shared__ float sew2p[32 * 32];   // paired [32,32]
  __shared__ float slwp[32 * 64];    // paired [32,64]
  __shared__ float slb[64];
  __shared__ float et[8][16 * 32];   // per-wave h tile, then e tile
  __shared__ int sidx[8][32];
  for (int i = threadIdx.x; i < 32; i += 256) { sew1[i] = ew1[i]; seb1[i] = eb1[i]; seb2[i] = eb2[i]; }
  stage_paired(ew2, sew2p, 32, 32, threadIdx.x, 256);
  stage_paired(lw, slwp, 32, 64, threadIdx.x, 256);
  for (int i = threadIdx.x; i < 64; i += 256) slb[i] = lb[i];
  __syncthreads();

  const int lane = threadIdx.x & 31;
  const int wave = threadIdx.x >> 5;
  const int ln = lane & 15;
  const int half = lane >> 4;
  float* etile = et[wave];
  int* idx = sidx[wave];
  const int e0 = (blockIdx.x * 8 + wave) * 16;

  if (lane < 16) {
    int e = e0 + lane; int ec = (e < ne) ? e : ne - 1;
    idx[lane] = src[ec];
    idx[16 + lane] = dst[ec];
  }
  { // hidden tile: lane covers edge ln, hidden cols [16*half, 16*half+16)
    int e = e0 + ln; int ec = (e < ne) ? e : ne - 1;
    float a = attr[ec];
#pragma unroll
    for (int c = 0; c < 16; ++c) {
      int j = half * 16 + c;
      etile[ln * 32 + j] = fmaxf(a * sew1[j] + seb1[j], 0.0f);
    }
  }
  __syncthreads();

  // ---- e = h @ ew2 + eb2 (no relu on edge MLP output) ----
  v8f eacc[2];
#pragma unroll
  for (int nt = 0; nt < 2; ++nt) {
    float bv = seb2[nt * 16 + ln];
#pragma unroll
    for (int v = 0; v < 8; ++v) eacc[nt][v] = bv;
  }
#pragma unroll
  for (int k = 0; k < 32; k += 4) {
    v2f a = *(const v2f*)&etile[ln * 32 + k + 2 * half];
#pragma unroll
    for (int nt = 0; nt < 2; ++nt)
      eacc[nt] = wmma4(a, bpair(sew2p, k, half, 32, nt * 16 + ln), eacc[nt]);
  }
  __syncthreads();
  // overwrite tile with e (all A reads above are complete)
#pragma unroll
  for (int nt = 0; nt < 2; ++nt)
#pragma unroll
    for (int v = 0; v < 8; ++v) {
      int m = v + 8 * half;
      etile[m * 32 + nt * 16 + ln] = eacc[nt][v];
    }
  __syncthreads();

  // ---- ee = e @ lin_w + lin_b ----
  v8f acc[4];
#pragma unroll
  for (int nt = 0; nt < 4; ++nt) {
    float bv = slb[nt * 16 + ln];
#pragma unroll
    for (int v = 0; v < 8; ++v) acc[nt][v] = bv;
  }
#pragma unroll
  for (int k = 0; k < 32; k += 4) {
    v2f a = *(const v2f*)&etile[ln * 32 + k + 2 * half];
#pragma unroll
    for (int nt = 0; nt < 4; ++nt)
      acc[nt] = wmma4(a, bpair(slwp, k, half, 64, nt * 16 + ln), acc[nt]);
  }
  // gather x1[src], relu, scatter-add to agg2[dst]
#pragma unroll
  for (int nt = 0; nt < 4; ++nt)
#pragma unroll
    for (int v = 0; v < 8; ++v) {
      int m = v + 8 * half;
      int e = e0 + m;
      if (e < ne) {
        int col = nt * 16 + ln;
        float mv = fmaxf(acc[nt][v] + x1[(size_t)idx[m] * 64 + col], 0.0f);
        atomicAdd(&agg2[(size_t)idx[16 + m] * 64 + col], mv);
      }
    }
}

// ----------------------------- pooling -------------------------------------
__global__ void pool_zero_kernel(float* __restrict__ sums, float* __restrict__ cnt, int g) {
  int i = blockIdx.x * blockDim.x + threadIdx.x;
  if (i < g * 64) sums[i] = 0.0f;
  if (i < g) cnt[i] = 0.0f;
}

__global__ void pool_acc_kernel(const float* __restrict__ x2, const int* __restrict__ batch,
                                float* __restrict__ sums, float* __restrict__ cnt, int n) {
  int gid = blockIdx.x * blockDim.x + threadIdx.x;
  int nid = gid >> 6, c = gid & 63;
  if (nid >= n) return;
  int g = batch[nid];
  atomicAdd(&sums[(size_t)g * 64 + c], x2[(size_t)nid * 64 + c]);
  if (c == 0) atomicAdd(&cnt[g], 1.0f);
}

// ----------------------------- regressor -----------------------------------
__global__ void final_kernel(const float* __restrict__ s_sums, const float* __restrict__ s_cnt,
                             const float* __restrict__ g_sums, const float* __restrict__ g_cnt,
                             const float* __restrict__ depth,
                             const float* __restrict__ w1, const float* __restrict__ b1, // [129,64]
                             const float* __restrict__ w2, const float* __restrict__ b2, // [64,1]
                             float* __restrict__ out, int ng) {
  int g = blockIdx.x * blockDim.x + threadIdx.x;
  if (g >= ng) return;
  float z[129];
  float sc = fmaxf(s_cnt[g], 1.0f), gc = fmaxf(g_cnt[g], 1.0f);
  for (int c = 0; c < 64; ++c) z[c] = s_sums[(size_t)g * 64 + c] / sc;
  for (int c = 0; c < 64; ++c) z[64 + c] = g_sums[(size_t)g * 64 + c] / gc;
  z[128] = depth[g];
  float o = b2[0];
  for (int jj = 0; jj < 64; ++jj) {
    float h = b1[jj];
    for (int i = 0; i < 129; ++i) h += z[i] * w1[i * 64 + jj];
    o += fmaxf(h, 0.0f) * w2[jj];
  }
  out[g] = o;
}

// ---------------------------------------------------------------------------
extern "C" void kernel_launch(void* const* d_in, const int* in_sizes, int n_in,
                              void* d_out, int out_size, void* d_ws, size_t ws_size,
                              hipStream_t stream) {
  const int N = in_sizes[0];        // nodes
  const int E = in_sizes[2];        // edges (edge_attr is [E,1])
  const int G = out_size;           // graphs

  const int* st_names = (const int*)d_in[0];
  const int* st_eidx  = (const int*)d_in[1];
  const float* st_attr = (const float*)d_in[2];
  const int* st_batch = (const int*)d_in[3];
  const int* go_names = (const int*)d_in[4];
  const int* go_eidx  = (const int*)d_in[5];
  const float* go_attr = (const float*)d_in[6];
  const int* go_batch = (const int*)d_in[7];
  const float* depth = (const float*)d_in[8];

  // params dict flattened in JAX-pytree (alphabetical key) order after input 8
#define FP(i) ((const float*)d_in[(i)])
  const int P = 9;
  const float* edge_b1 = FP(P + 0);
  const float* edge_b2 = FP(P + 1);
  const float* edge_w1 = FP(P + 2);
  const float* edge_w2 = FP(P + 3);
  const float* gc1_b1 = FP(P + 4);
  const float* gc1_b2 = FP(P + 5);
  const float* gc1_w1 = FP(P + 6);
  const float* gc1_w2 = FP(P + 7);
  const float* gc2_b1 = FP(P + 8);
  const float* gc2_b2 = FP(P + 9);
  const float* gc2_lin_b = FP(P + 10);
  const float* gc2_lin_w = FP(P + 11);
  const float* gc2_w1 = FP(P + 12);
  const float* gc2_w2 = FP(P + 13);
  const float* id_b1 = FP(P + 14);
  const float* id_b2 = FP(P + 15);
  const float* id_w1 = FP(P + 16);
  const float* id_w2 = FP(P + 17);
  const float* reg_b1 = FP(P + 18);
  const float* reg_b2 = FP(P + 19);
  const float* reg_w1 = FP(P + 20);
  const float* reg_w2 = FP(P + 21);
  const float* sc1_b1 = FP(P + 22);
  const float* sc1_b2 = FP(P + 23);
  const float* sc1_w1 = FP(P + 24);
  const float* sc1_w2 = FP(P + 25);
  const float* sc2_b1 = FP(P + 26);
  const float* sc2_b2 = FP(P + 27);
  const float* sc2_lin_b = FP(P + 28);
  const float* sc2_lin_w = FP(P + 29);
  const float* sc2_w1 = FP(P + 30);
  const float* sc2_w2 = FP(P + 31);
#undef FP

  // workspace layout (fp32)
  float* ws = (float*)d_ws;
  float* x0 = ws;                             // N*32
  float* agg1 = x0 + (size_t)N * 32;          // N*32
  float* x1 = agg1 + (size_t)N * 32;          // N*64
  float* agg2 = x1 + (size_t)N * 64;          // N*64
  float* x2 = agg2 + (size_t)N * 64;          // N*64
  float* s_sums = x2 + (size_t)N * 64;        // G*64
  float* s_cnt = s_sums + (size_t)G * 64;     // G
  float* g_sums = s_cnt + G;                  // G*64
  float* g_cnt = g_sums + (size_t)G * 64;     // G

  const int nodeBlocks = (N + 255) / 256;
  const int tileNodeBlocks = (N + 127) / 128; // 8 waves * 16 rows
  const int tileEdgeBlocks = (E + 127) / 128;
  const int poolBlocks = (int)(((size_t)N * 64 + 255) / 256);
  const int zeroBlocks = (G * 64 + 255) / 256;

  auto encode = [&](const int* names, const int* eidx, const float* attr, const int* batch,
                    const float* c1w1, const float* c1b1, const float* c1w2, const float* c1b2,
                    const float* c2w1, const float* c2b1, const float* c2w2, const float* c2b2,
                    const float* c2lw, const float* c2lb,
                    float* sums, float* cnt) {
    const int* src = eidx;
    const int* dst = eidx + E;
    node_embed_kernel<<<nodeBlocks, 256, 0, stream>>>(
        names, id_w1, id_b1, id_w2, id_b2, x0, agg1, N);
    conv1_msg_kernel<<<tileEdgeBlocks, 256, 0, stream>>>(
        src, dst, attr, edge_w1, edge_b1, edge_w2, edge_b2, x0, agg1, E);
    node_mlp_kernel<32><<<tileNodeBlocks, 256, 0, stream>>>(
        agg1, c1w1, c1b1, c1w2, c1b2, x1, agg2, N);
    conv2_msg_kernel<<<tileEdgeBlocks, 256, 0, stream>>>(
        src, dst, attr, edge_w1, edge_b1, edge_w2, edge_b2, c2lw, c2lb, x1, agg2, E);
    node_mlp_kernel<64><<<tileNodeBlocks, 256, 0, stream>>>(
        agg2, c2w1, c2b1, c2w2, c2b2, x2, nullptr, N);
    pool_zero_kernel<<<zeroBlocks, 256, 0, stream>>>(sums, cnt, G);
    pool_acc_kernel<<<poolBlocks, 256, 0, stream>>>(x2, batch, sums, cnt, N);
  };

  encode(st_names, st_eidx, st_attr, st_batch,
         sc1_w1, sc1_b1, sc1_w2, sc1_b2,
         sc2_w1, sc2_b1, sc2_w2, sc2_b2, sc2_lin_w, sc2_lin_b,
         s_sums, s_cnt);
  encode(go_names, go_eidx, go_attr, go_batch,
         gc1_w1, gc1_b1, gc1_w2, gc1_b2,
         gc2_w1, gc2_b1, gc2_w2, gc2_b2, gc2_lin_w, gc2_lin_b,
         g_sums, g_cnt);

  final_kernel<<<(G + 255) / 256, 256, 0, stream>>>(
      s_sums, s_cnt, g_sums, g_cnt, depth,
      reg_w1, reg_b1, reg_w2, reg_b2, (float*)d_out, G);

  (void)n_in; (void)ws_size;
}